// GCN_309237645840
// MI455X (gfx1250) — compile-verified
//
#include <hip/hip_runtime.h>
#include <math.h>

typedef __attribute__((ext_vector_type(2))) float v2f;
typedef __attribute__((ext_vector_type(8))) float v8f;

#define IN_F 256
#define OUT_F 128
#define KT_TILES (IN_F / 4)    // 64 k-steps of 4
#define NT_TILES (OUT_F / 16)  // 8 n-tiles of 16
#define LDS_STRIDE (IN_F + 1)  // 257: break 64-bank conflicts on fragment reads

// ---------------------------------------------------------------------------
// Kernel 0: pack W[128][256] (row-major) into WMMA B-fragment order so the
// GEMM's per-wave B loads are a single fully-coalesced float2 per lane.
// Element for (kt, nt, lane): n = nt*16 + (lane&15), kb = kt*4 + 2*(lane>>4)
// value = { W[n][kb], W[n][kb+1] }  (i.e. B = W^T loaded column-major)
// ---------------------------------------------------------------------------
__global__ void pack_w_kernel(const float* __restrict__ W, float* __restrict__ Bp) {
  int tid = blockIdx.x * blockDim.x + threadIdx.x;
  if (tid >= KT_TILES * NT_TILES * 32) return;
  int lane = tid & 31;
  int t    = tid >> 5;
  int nt   = t % NT_TILES;
  int kt   = t / NT_TILES;
  int n    = nt * 16 + (lane & 15);
  int kb   = kt * 4 + 2 * (lane >> 4);
  Bp[2 * tid]     = W[n * IN_F + kb];
  Bp[2 * tid + 1] = W[n * IN_F + kb + 1];
}

// ---------------------------------------------------------------------------
// Kernel 1: fts = tanh(seq @ W^T)   [exact f32 via V_WMMA_F32_16X16X4_F32]
// grid = N/16 blocks of 256 threads (8 waves). Block stages a 16x256 slice of
// seq in LDS; wave w owns n-tile w and accumulates a 16x16 f32 tile in 8 VGPRs.
// ---------------------------------------------------------------------------
__global__ void __launch_bounds__(256)
gemm_tanh_kernel(const float* __restrict__ seq, const float* __restrict__ Bp,
                 const int* __restrict__ active, float* __restrict__ fts) {
  __shared__ float As[16 * LDS_STRIDE];
  const int m0 = blockIdx.x * 16;
  const int t  = threadIdx.x;

  // cooperative, coalesced load of the 16x256 A block (one full row per pass)
  #pragma unroll
  for (int i = 0; i < 16; ++i)
    As[i * LDS_STRIDE + t] = seq[(size_t)(m0 + i) * IN_F + t];
  __syncthreads();

  const int lane = t & 31;
  const int wv   = t >> 5;        // n-tile owned by this wave (0..7)
  const int m    = lane & 15;     // A row / B column within tile
  const int hi   = lane >> 4;     // K-half selector

  const v2f* __restrict__ bptr = (const v2f*)Bp + (size_t)wv * 32 + lane;

  v8f acc = {};
  #pragma unroll 4
  for (int kt = 0; kt < KT_TILES; ++kt) {
    const int kb = kt * 4 + 2 * hi;
    v2f a;
    a.x = As[m * LDS_STRIDE + kb];
    a.y = As[m * LDS_STRIDE + kb + 1];
    v2f b = bptr[(size_t)kt * (NT_TILES * 32)];  // coalesced 256B per wave
    // D = A(16x4,f32) x B(4x16,f32) + C(16x16,f32)
    acc = __builtin_amdgcn_wmma_f32_16x16x4_f32(
        /*neg_a=*/false, a, /*neg_b=*/false, b,
        /*c_mod=*/(short)0, acc, /*reuse_a=*/false, /*reuse_b=*/false);
  }

  const bool act = (*active != 0);   // uniform branch, after all WMMAs
  const int  n   = wv * 16 + m;
  #pragma unroll
  for (int r = 0; r < 8; ++r) {
    float v = acc[r];
    if (act) v = tanhf(v);
    const int M = m0 + r + 8 * hi;   // D layout: VGPR r -> row r (+8 for hi half)
    fts[(size_t)M * OUT_F + n] = v;
  }
}

// ---------------------------------------------------------------------------
// Kernel 2: CSR row_start[i] = lower_bound(edge_row, i)  (edge_row is sorted)
// ---------------------------------------------------------------------------
__global__ void rowptr_kernel(const int* __restrict__ edge_row, int E, int N,
                              int* __restrict__ row_start) {
  int i = blockIdx.x * blockDim.x + threadIdx.x;
  if (i > N) return;
  int lo = 0, hi = E;
  while (lo < hi) {
    int mid = (lo + hi) >> 1;
    if (edge_row[mid] < i) lo = mid + 1; else hi = mid;
  }
  row_start[i] = lo;
}

// ---------------------------------------------------------------------------
// Kernel 3: deterministic segment-sum SpMM + bias.
// One block per node, one thread per feature; gathers hit L2 (fts = 51 MB).
// ---------------------------------------------------------------------------
__global__ void __launch_bounds__(OUT_F)
spmm_kernel(const float* __restrict__ fts, const float* __restrict__ ew,
            const int* __restrict__ ecol, const int* __restrict__ row_start,
            const float* __restrict__ bias, float* __restrict__ out) {
  const int i = blockIdx.x;
  const int f = threadIdx.x;
  const int s = row_start[i];
  const int e = row_start[i + 1];
  float a0 = 0.f, a1 = 0.f;
  int j = s;
  for (; j + 1 < e; j += 2) {         // 2-way unroll for latency hiding
    const int   c0 = ecol[j],   c1 = ecol[j + 1];
    const float w0 = ew[j],     w1 = ew[j + 1];
    a0 = fmaf(w0, fts[(size_t)c0 * OUT_F + f], a0);
    a1 = fmaf(w1, fts[(size_t)c1 * OUT_F + f], a1);
  }
  if (j < e)
    a0 = fmaf(ew[j], fts[(size_t)ecol[j] * OUT_F + f], a0);
  out[(size_t)i * OUT_F + f] = a0 + a1 + bias[f];
}

// ---------------------------------------------------------------------------
extern "C" void kernel_launch(void* const* d_in, const int* in_sizes, int n_in,
                              void* d_out, int out_size, void* d_ws, size_t ws_size,
                              hipStream_t stream) {
  (void)n_in; (void)out_size; (void)ws_size;
  const float* seq    = (const float*)d_in[0];
  const float* W      = (const float*)d_in[1];
  const float* bias   = (const float*)d_in[2];
  const float* ew     = (const float*)d_in[3];
  const int*   erow   = (const int*)d_in[4];
  const int*   ecol   = (const int*)d_in[5];
  const int*   active = (const int*)d_in[6];

  const int N = in_sizes[0] / IN_F;   // 100000 (divisible by 16)
  const int E = in_sizes[3];          // 3200000

  // workspace layout: fts | Bpack | row_start
  char*  ws  = (char*)d_ws;
  float* fts = (float*)ws;
  size_t off = (size_t)N * OUT_F * sizeof(float);
  off = (off + 255) & ~(size_t)255;
  float* Bp = (float*)(ws + off);
  off += (size_t)KT_TILES * NT_TILES * 32 * 2 * sizeof(float);
  off = (off + 255) & ~(size_t)255;
  int* row_start = (int*)(ws + off);

  pack_w_kernel<<<(KT_TILES * NT_TILES * 32 + 255) / 256, 256, 0, stream>>>(W, Bp);
  rowptr_kernel<<<(N + 1 + 255) / 256, 256, 0, stream>>>(erow, E, N, row_start);
  gemm_tanh_kernel<<<N / 16, 256, 0, stream>>>(seq, Bp, active, fts);
  spmm_kernel<<<N, OUT_F, 0, stream>>>(fts, ew, ecol, row_start, bias, (float*)d_out);
}